// RoIAlignAvg_3d_23252952941137
// MI455X (gfx1250) — compile-verified
//
#include <hip/hip_runtime.h>
#include <hip/hip_bf16.h>

typedef __attribute__((ext_vector_type(2))) float v2f;
typedef __attribute__((ext_vector_type(8))) float v8f;

#define NS 16           // samples per axis = pooled(8) * sampling_ratio(2)
#define NC 64
#define DD 48
#define HH 128
#define WW 128
#define OUTD 7          // 7x7x7 output per (roi, channel)

// Reduction matrix entry: out o accumulates samples s in [2o, 2o+3].
__device__ __forceinline__ float redw(int s, int o) {
  return (o < 7 && s >= 2 * o && s <= 2 * o + 3) ? 1.0f : 0.0f;
}

__global__ __launch_bounds__(128) void roialign_avg3d_kernel(
    const float* __restrict__ feat, const float* __restrict__ rois,
    float* __restrict__ out) {
  const int c = blockIdx.x;    // channel
  const int r = blockIdx.y;    // roi
  const int t = threadIdx.x;   // 128 threads = 4 waves (wave32)
  const int lane = t & 31;
  const int wv = t >> 5;

  __shared__ float s_samp[NS][NS][NS];   // 16 KB sample cube (reused for U)
  __shared__ float s_tmp[NS][NS][NS];    // 16 KB x-reduced slices
  __shared__ int   s_lo[3][NS];
  __shared__ int   s_hi[3][NS];
  __shared__ float s_fr[3][NS];
  __shared__ float s_vd[3][NS];
  __shared__ float s_roi[7];

  if (t < 7) s_roi[t] = rois[r * 7 + t];
  __syncthreads();

  // ---- Stage 0: per-axis coordinate tables (threads 0..47) ----
  if (t < 48) {
    const int axis = t >> 4;   // 0=x(W), 1=y(H), 2=z(D)
    const int i = t & 15;
    const float scale = 0.25f;
    float start, stop, dimf;
    if (axis == 0)      { start = s_roi[1] * scale; stop = s_roi[4] * scale; dimf = (float)WW; }
    else if (axis == 1) { start = s_roi[2] * scale; stop = s_roi[5] * scale; dimf = (float)HH; }
    else                { start = s_roi[3] * scale; stop = s_roi[6] * scale; dimf = (float)DD; }
    float len = fmaxf(stop - start, 1.0f);
    float bin = len * (1.0f / 8.0f);
    float cc = start + (float)(i >> 1) * bin + ((float)(i & 1) + 0.5f) * bin * 0.5f;
    float valid = (cc >= -1.0f && cc <= dimf) ? 1.0f : 0.0f;
    cc = fminf(fmaxf(cc, 0.0f), dimf - 1.0f);
    float lo = floorf(cc);
    float hi = fminf(lo + 1.0f, dimf - 1.0f);
    s_lo[axis][i] = (int)lo;
    s_hi[axis][i] = (int)hi;
    s_fr[axis][i] = cc - lo;
    s_vd[axis][i] = valid;
  }
  __syncthreads();

  // ---- Stage 1: gather 16^3 trilinear samples into LDS ----
  const int b = (int)s_roi[0];
  const float* fc = feat + ((size_t)b * NC + c) * (size_t)(DD * HH * WW);

  for (int it = 0; it < 32; ++it) {
    const int idx = t + it * 128;
    const int xs = idx & 15;
    const int ys = (idx >> 4) & 15;
    const int zs = idx >> 8;

    const int x0 = s_lo[0][xs], xh = s_hi[0][xs];
    const int y0 = s_lo[1][ys], yh = s_hi[1][ys];
    const int z0 = s_lo[2][zs], zh = s_hi[2][zs];
    const float fx = s_fr[0][xs], fy = s_fr[1][ys], fz = s_fr[2][zs];
    const float vflag = s_vd[0][xs] * s_vd[1][ys] * s_vd[2][zs];

    const float* p00 = fc + (size_t)z0 * (HH * WW) + (size_t)y0 * WW;
    const float* p01 = fc + (size_t)z0 * (HH * WW) + (size_t)yh * WW;
    const float* p10 = fc + (size_t)zh * (HH * WW) + (size_t)y0 * WW;
    const float* p11 = fc + (size_t)zh * (HH * WW) + (size_t)yh * WW;

    // Hint the far-z rows toward L2 while near-z corners are in flight.
    __builtin_prefetch(p10 + x0, 0, 1);
    __builtin_prefetch(p11 + x0, 0, 1);

    const float v000 = p00[x0], v001 = p00[xh];
    const float v010 = p01[x0], v011 = p01[xh];
    const float v100 = p10[x0], v101 = p10[xh];
    const float v110 = p11[x0], v111 = p11[xh];

    const float wx0 = 1.0f - fx, wy0 = 1.0f - fy, wz0 = 1.0f - fz;
    const float s =
        wz0 * (wy0 * (wx0 * v000 + fx * v001) + fy * (wx0 * v010 + fx * v011)) +
        fz  * (wy0 * (wx0 * v100 + fx * v101) + fy * (wx0 * v110 + fx * v111));
    s_samp[zs][ys][xs] = s * vflag;
  }
  __syncthreads();

  // WMMA fragment coordinates (fp32 16x16x4, wave32):
  //   A (16x4): lanes 0-15 hold M=lane {VGPR0:K+0, VGPR1:K+1};
  //             lanes 16-31 hold M=lane-16 {VGPR0:K+2, VGPR1:K+3}.
  //   B (4x16): same striping with N instead of M.
  //   C/D (16x16): VGPR v = row M=v (lanes 0-15) / M=v+8 (lanes 16-31), N=lane&15.
  const int m = lane & 15;
  const int kh = lane >> 4;     // 0 or 1 -> K offset 0 or 2

  // ---- Stage 2: x-reduction per z-slice: T[z][y][ox] = sum_x S[z][y][x]*Rx[x][ox]
  for (int s = 0; s < 4; ++s) {
    const int z = (wv << 2) | s;   // each wave owns 4 slices
    v8f acc = {};
    for (int kk = 0; kk < 16; kk += 4) {
      v2f a, bm;
      a.x = s_samp[z][m][kk + 2 * kh + 0];
      a.y = s_samp[z][m][kk + 2 * kh + 1];
      bm.x = redw(kk + 2 * kh + 0, m);
      bm.y = redw(kk + 2 * kh + 1, m);
      acc = __builtin_amdgcn_wmma_f32_16x16x4_f32(
          false, a, false, bm, (short)0, acc, false, false);
    }
    for (int v = 0; v < 8; ++v)
      s_tmp[z][v + 8 * kh][m] = acc[v];
  }
  __syncthreads();

  // ---- Stage 3: y-reduction per z-slice: U[z][oy][ox] = sum_y Ry^T[oy][y]*T[z][y][ox]
  for (int s = 0; s < 4; ++s) {
    const int z = (wv << 2) | s;
    v8f acc = {};
    for (int kk = 0; kk < 16; kk += 4) {
      v2f a, bm;
      a.x = redw(kk + 2 * kh + 0, m);   // Ry^T[m][k] = redw(k, m)
      a.y = redw(kk + 2 * kh + 1, m);
      bm.x = s_tmp[z][kk + 2 * kh + 0][m];
      bm.y = s_tmp[z][kk + 2 * kh + 1][m];
      acc = __builtin_amdgcn_wmma_f32_16x16x4_f32(
          false, a, false, bm, (short)0, acc, false, false);
    }
    for (int v = 0; v < 8; ++v)
      s_samp[z][v + 8 * kh][m] = acc[v];   // reuse as U[z][oy][ox]
  }
  __syncthreads();

  // ---- Stage 4: z-reduction + 1/64 scale, write 7x7x7 outputs ----
  float* outp = out + ((size_t)r * NC + c) * (OUTD * OUTD * OUTD);
  for (int i = t; i < OUTD * OUTD * OUTD; i += 128) {
    const int oz = i / 49;
    const int rem = i % 49;
    const int oy = rem / 7;
    const int ox = rem % 7;
    float acc = 0.0f;
    for (int dz = 0; dz < 4; ++dz)
      acc += s_samp[2 * oz + dz][oy][ox];
    outp[i] = acc * (1.0f / 64.0f);
  }
}

extern "C" void kernel_launch(void* const* d_in, const int* in_sizes, int n_in,
                              void* d_out, int out_size, void* d_ws, size_t ws_size,
                              hipStream_t stream) {
  const float* features = (const float*)d_in[0];   // (2, 64, 48, 128, 128) f32
  const float* rois = (const float*)d_in[1];       // (64, 7) f32
  float* out = (float*)d_out;                      // (64, 64, 7, 7, 7) f32
  (void)in_sizes; (void)n_in; (void)out_size; (void)d_ws; (void)ws_size;

  dim3 grid(NC, 64);   // x = channel, y = roi
  dim3 block(128);
  roialign_avg3d_kernel<<<grid, block, 0, stream>>>(features, rois, out);
}